// MoELayerStacks_14121852469539
// MI455X (gfx1250) — compile-verified
//
#include <hip/hip_runtime.h>
#include <math.h>

// ---------------------------------------------------------------------------
// MoE layer stack for MI455X (gfx1250, wave32).
//
// Roofline: x is 16384x3072 fp32 = 192 MB -> 8.3 us at 23.3 TB/s HBM. Routing
// FIRST cuts the big einsum 8x (only selected expert computed): 0.8 G-MAC
// fp32, far below the fp32 WMMA ceiling in 8 us -> memory bound, so we keep
// full fp32 precision and use V_WMMA_F32_16X16X4_F32 for the 3072-deep GEMM.
// ---------------------------------------------------------------------------

typedef __attribute__((ext_vector_type(2))) float v2f;
typedef __attribute__((ext_vector_type(8))) float v8f;

#define NB      16384      // batch rows
#define NL1     3072       // input features
#define NL2     15         // hidden (pre-duplication)
#define NL3     32         // second hidden
#define NE      8          // experts
#define NRF     32         // router feats per half

// ---------------------------------------------------------------------------
// Kernel 1: router + per-expert row compaction.
// ws layout: [0,32) int counts[NE] (padded to 256B), then rowlist[NE][NB].
// ---------------------------------------------------------------------------
__global__ __launch_bounds__(256) void moe_router_kernel(
    const float* __restrict__ x,
    const float* __restrict__ rw,   // (E, 64)
    const float* __restrict__ rb,   // (E,)
    int* __restrict__ counts,
    int* __restrict__ rowlist)
{
    __shared__ float s_rw[NE][2 * NRF];
    __shared__ float s_rb[NE];
    const int t = threadIdx.x;
    for (int i = t; i < NE * 2 * NRF; i += 256) ((float*)s_rw)[i] = rw[i];
    if (t < NE) s_rb[t] = rb[t];
    __syncthreads();

    const int b = blockIdx.x * 256 + t;
    const float* xr = x + (size_t)b * NL1;

    float xa[2 * NRF];
#pragma unroll
    for (int i = 0; i < NRF; ++i) xa[i] = xr[i];
#pragma unroll
    for (int i = 0; i < NRF; ++i) xa[NRF + i] = xr[NL1 / 2 + i];

    float best = -__builtin_inff();
    int bi = 0;
#pragma unroll
    for (int e = 0; e < NE; ++e) {
        float s = s_rb[e];
#pragma unroll
        for (int i = 0; i < 2 * NRF; ++i) s += xa[i] * s_rw[e][i];
        if (s > best) { best = s; bi = e; }   // strict > == first-max (jnp.argmax)
    }
    const int slot = atomicAdd(&counts[bi], 1);
    rowlist[bi * NB + slot] = b;
}

// ---------------------------------------------------------------------------
// Kernel 2: per-expert gathered GEMM (WMMA f32 16x16x4) + tiny MLP tail.
// grid = (128, NE); block = 256 threads = 8 wave32s; 1 wave per 16-row tile.
// ---------------------------------------------------------------------------
__global__ __launch_bounds__(256) void moe_expert_kernel(
    const float* __restrict__ x,
    const float* __restrict__ l1w,  // (E, 16, 3072)
    const float* __restrict__ l1b,  // (E, 16)
    const float* __restrict__ l2w,  // (E, 32, 30)
    const float* __restrict__ l2b,  // (E, 32)
    const float* __restrict__ ow,   // (E, 1, 32)
    const float* __restrict__ ob,   // (E, 1)
    const int* __restrict__ counts,
    const int* __restrict__ rowlist,
    float* __restrict__ out)
{
    const int e = blockIdx.y;
    const int count = counts[e];
    const int tilesTotal = (count + 15) >> 4;
    const int blockFirstTile = blockIdx.x * 8;
    if (blockFirstTile >= tilesTotal) return;   // block-uniform: no barrier issues

    __shared__ float s_tile[8][16][17];         // per-wave D tile, padded
    __shared__ float s_l2w[NL3][2 * NL2];
    __shared__ float s_l2b[NL3];
    __shared__ float s_ow[NL3];
    __shared__ float s_l1b[16];

    const int t = threadIdx.x;
    for (int i = t; i < NL3 * 2 * NL2; i += 256)
        ((float*)s_l2w)[i] = l2w[e * NL3 * 2 * NL2 + i];
    if (t < NL3) { s_l2b[t] = l2b[e * NL3 + t]; s_ow[t] = ow[e * NL3 + t]; }
    if (t < 16)  s_l1b[t] = l1b[e * 16 + t];
    __syncthreads();

    const int wave = t >> 5;
    const int lane = t & 31;
    const int m    = lane & 15;   // A-row / B-col / local tile row
    const int h    = lane >> 4;   // K-half selector (ISA 32-bit A/B layout)
    const int tile = blockFirstTile + wave;
    const bool tvalid = tile < tilesTotal;      // wave-uniform -> EXEC all-1s in WMMA

    v8f acc = {};
    int row = 0;
    if (tvalid) {
        int gidx = tile * 16 + m;
        if (gidx > count - 1) gidx = count - 1;         // pad partial tile (dup row)
        row = rowlist[e * NB + gidx];

        // A[m][kk] = x[row_m][k+kk]; lane holds (k+2h, k+2h+1) in v[0],v[1].
        const float* aPtr = x  + (size_t)row * NL1 + 2 * h;
        // B[kk][n] = l1_w[e][n][k+kk]; lane n==m, same K-half striping.
        const float* bPtr = l1w + ((size_t)e * 16 + m) * NL1 + 2 * h;

        for (int k0 = 0; k0 < NL1; k0 += 64) {
            // stream the gathered row ahead (global_prefetch_b8)
            if (k0 + 256 < NL1) __builtin_prefetch(aPtr + k0 + 256, 0, 0);
#pragma unroll
            for (int kk = 0; kk < 64; kk += 4) {
                v2f a = *(const v2f*)(aPtr + k0 + kk);
                v2f b = *(const v2f*)(bPtr + k0 + kk);
                acc = __builtin_amdgcn_wmma_f32_16x16x4_f32(
                    false, a, false, b, (short)0, acc, false, false);
            }
        }
        // D layout: VGPR v, lanes0-15 -> M=v, lanes16-31 -> M=v+8; N = lane&15.
#pragma unroll
        for (int v = 0; v < 8; ++v) s_tile[wave][v + 8 * h][m] = acc[v];
    }
    __syncthreads();

    if (tvalid) {
        const int r = m;   // local row handled by this lane
        const int p = h;   // which half of the 32 l2-outputs

        float l1v[16];
#pragma unroll
        for (int n = 0; n < 16; ++n) l1v[n] = s_tile[wave][r][n] + s_l1b[n];

        // features: [clip(v^2*255/256), clip(v)] for the first 15 channels
        float feat[2 * NL2];
#pragma unroll
        for (int j = 0; j < NL2; ++j) {
            float q = l1v[j] * l1v[j] * (255.0f / 256.0f);
            feat[j]       = fminf(fmaxf(q, 0.0f), 1.0f);
            feat[NL2 + j] = fminf(fmaxf(l1v[j], 0.0f), 1.0f);
        }

        float partial = 0.0f;
#pragma unroll
        for (int tt = 0; tt < 16; ++tt) {
            const int o = p * 16 + tt;
            float s = s_l2b[o];
#pragma unroll
            for (int j = 0; j < 2 * NL2; ++j) s += feat[j] * s_l2w[o][j];
            s = fminf(fmaxf(s, 0.0f), 1.0f);
            partial += s * s_ow[o];
        }
        // wave32 butterfly: combine the two half-dots
        const float total = partial + __shfl_xor(partial, 16, 32);

        const int g = tile * 16 + r;
        if (p == 0 && g < count)
            out[row] = total + ob[e] + l1v[15];   // + skip connection l1x_out
    }
}

// ---------------------------------------------------------------------------
extern "C" void kernel_launch(void* const* d_in, const int* in_sizes, int n_in,
                              void* d_out, int out_size, void* d_ws, size_t ws_size,
                              hipStream_t stream) {
    const float* x   = (const float*)d_in[0];
    const float* rw  = (const float*)d_in[1];
    const float* rb  = (const float*)d_in[2];
    const float* l1w = (const float*)d_in[3];
    const float* l1b = (const float*)d_in[4];
    const float* l2w = (const float*)d_in[5];
    const float* l2b = (const float*)d_in[6];
    const float* ow  = (const float*)d_in[7];
    const float* ob  = (const float*)d_in[8];
    float* out = (float*)d_out;

    int* counts  = (int*)d_ws;
    int* rowlist = (int*)((char*)d_ws + 256);

    hipMemsetAsync(counts, 0, 256, stream);
    moe_router_kernel<<<NB / 256, 256, 0, stream>>>(x, rw, rb, counts, rowlist);

    dim3 grid(128, NE);   // 128 blocks x 8 waves x 16 rows = 16384 worst case/expert
    moe_expert_kernel<<<grid, 256, 0, stream>>>(x, l1w, l1b, l2w, l2b, ow, ob,
                                                counts, rowlist, out);
}